// Multi_Loss_89129161326919
// MI455X (gfx1250) — compile-verified
//
#include <hip/hip_runtime.h>
#include <hip/hip_bf16.h>

#define EPSF 1e-10f

typedef __attribute__((ext_vector_type(2))) float v2f;
typedef __attribute__((ext_vector_type(8))) float v8f;

__device__ __forceinline__ float softplus_eps(float x) {
    // reference: log(1 + exp(x) + 1e-10)
    return __logf(1.0f + __expf(x) + EPSF);
}

__device__ __forceinline__ float wave_reduce_sum(float v) {
#pragma unroll
    for (int off = 16; off > 0; off >>= 1) v += __shfl_xor(v, off, 32);
    return v;
}

// blockDim.x == 256 (8 wave32's); returns the sum to ALL threads.
__device__ __forceinline__ float block_reduce_sum(float v, float* red) {
    v = wave_reduce_sum(v);
    const int wid = threadIdx.x >> 5;
    if ((threadIdx.x & 31) == 0) red[wid] = v;
    __syncthreads();
    float s = 0.0f;
    if (threadIdx.x == 0) {
#pragma unroll
        for (int i = 0; i < 8; ++i) s += red[i];
        red[0] = s;
    }
    __syncthreads();
    s = red[0];
    __syncthreads();
    return s;
}

__global__ __launch_bounds__(256) void multi_loss_kernel(
    const float* __restrict__ wacc_p, const float* __restrict__ wdiv_p,
    const float* __restrict__ wfair_p,
    const int* __restrict__ pos_items,
    const float* __restrict__ all_scores,
    const float* __restrict__ pos_scores, const float* __restrict__ neg_scores,
    const float* __restrict__ igm,
    float* __restrict__ out, int P, int N, int G)
{
    __shared__ float red[8];
    __shared__ int   s_items[64];          // P <= 64
    const int b = blockIdx.x;
    const int t = threadIdx.x;

    // Stage this row's item indices into LDS (coalesced, once). Visibility for
    // phase D is guaranteed by the __syncthreads inside the block reductions.
    if (t < P) s_items[t] = pos_items[b * P + t];

    const float*  rowf = all_scores + (size_t)b * (size_t)N;
    const float4* row4 = (const float4*)rowf;
    const int nv = N >> 2;

    // ---------- Fairness pass 1: mean of softplus(all_scores[b,:]) ----------
    float lsum = 0.0f;
    for (int i = t; i < nv; i += 256) {
        __builtin_prefetch((const void*)(row4 + i + 512), 0, 1); // global_prefetch_b8
        float4 v = row4[i];
        lsum += softplus_eps(v.x) + softplus_eps(v.y) +
                softplus_eps(v.z) + softplus_eps(v.w);
    }
    for (int i = (nv << 2) + t; i < N; i += 256) lsum += softplus_eps(rowf[i]);
    const float mean = block_reduce_sum(lsum, red) / (float)N;

    // ---------- Fairness pass 2: sum |softplus - mean| (row is L2-resident) ----------
    float lfair = 0.0f;
    for (int i = t; i < nv; i += 256) {
        float4 v = row4[i];
        lfair += fabsf(softplus_eps(v.x) - mean) + fabsf(softplus_eps(v.y) - mean)
               + fabsf(softplus_eps(v.z) - mean) + fabsf(softplus_eps(v.w) - mean);
    }
    for (int i = (nv << 2) + t; i < N; i += 256)
        lfair += fabsf(softplus_eps(rowf[i]) - mean);
    const float fair = block_reduce_sum(lfair, red);

    // ---------- Accuracy: -sum log(eps + sigmoid(pos - neg)) ----------
    float lacc = 0.0f;
    for (int p = t; p < P; p += 256) {
        float d = pos_scores[b * P + p] - neg_scores[b * P + p];
        float s = 1.0f / (1.0f + __expf(-d));
        lacc -= __logf(EPSF + s);
    }
    const float acc = block_reduce_sum(lacc, red);

    // ---------- Diversity: genre entropy (wave 0 only; lane l owns genre l) ----------
    float divloss = 0.0f;
    if (t < 32) {
        const int l = t;
        float gc;
#if __has_builtin(__builtin_amdgcn_wmma_f32_16x16x4_f32)
        {
            // D = A(16x4, all ones) * B(4x16 gathered genre rows) + C
            // => every row of D is the column sum over the 4 items fed as B.
            const int g = l & 15;
            const int lo = (l < 16) ? 0 : 2;
            v2f a; a.x = 1.0f; a.y = 1.0f;
            v8f c0 = {}; v8f c1 = {};
            const int full = P >> 2;               // 12 unpredicated chunks
            for (int j = 0; j < full; ++j) {
                const int p0 = 4 * j + lo;
                const int r0 = s_items[p0]     * G + g;
                const int r1 = s_items[p0 + 1] * G + g;
                v2f bl, bh;
                bl.x = igm[r0];      bl.y = igm[r1];
                bh.x = igm[r0 + 16]; bh.y = igm[r1 + 16];
                c0 = __builtin_amdgcn_wmma_f32_16x16x4_f32(
                        false, a, false, bl, (short)0, c0, false, false);
                c1 = __builtin_amdgcn_wmma_f32_16x16x4_f32(
                        false, a, false, bh, (short)0, c1, false, false);
            }
            if (P & 3) {                           // tail chunk, branch-free
                const int p0 = 4 * full + lo;
                const int p1 = p0 + 1;
                const float m0 = (p0 < P) ? 1.0f : 0.0f;
                const float m1 = (p1 < P) ? 1.0f : 0.0f;
                const int q0 = (p0 < P) ? p0 : (P - 1);
                const int q1 = (p1 < P) ? p1 : (P - 1);
                const int r0 = s_items[q0] * G + g;
                const int r1 = s_items[q1] * G + g;
                v2f bl, bh;
                bl.x = m0 * igm[r0];      bl.y = m1 * igm[r1];
                bh.x = m0 * igm[r0 + 16]; bh.y = m1 * igm[r1 + 16];
                c0 = __builtin_amdgcn_wmma_f32_16x16x4_f32(
                        false, a, false, bl, (short)0, c0, false, false);
                c1 = __builtin_amdgcn_wmma_f32_16x16x4_f32(
                        false, a, false, bh, (short)0, c1, false, false);
            }
            // C/D layout: VGPR0, lanes 0-15 -> (M=0, N=lane); lanes 16-31 -> (M=8, N=lane-16).
            // All M rows identical (ones-A), so lane l reads genre l's count:
            gc = (l < 16) ? c0[0] : c1[0];
        }
#else
        {
            gc = 0.0f;
            for (int p = 0; p < P; ++p) gc += igm[s_items[p] * G + l];
        }
#endif
        const float denom = wave_reduce_sum(gc);
        float prob = (denom > 0.0f) ? (gc / denom) : 0.0f;
        prob = fmaxf(prob, EPSF);                       // jnp.clip(p, 1e-10, None)
        divloss = wave_reduce_sum(-prob * __logf(prob + EPSF));
    }

    if (t == 0) {
        out[b] = wacc_p[0] * acc + wdiv_p[0] * divloss + wfair_p[0] * fair;
    }
}

extern "C" void kernel_launch(void* const* d_in, const int* in_sizes, int n_in,
                              void* d_out, int out_size, void* d_ws, size_t ws_size,
                              hipStream_t stream) {
    const float* wacc       = (const float*)d_in[0];
    const float* wdiv       = (const float*)d_in[1];
    const float* wfair      = (const float*)d_in[2];
    const int*   pos_items  = (const int*)  d_in[3];
    // d_in[4] = neg_items: unused by the reference math (it uses neg_scores)
    const float* all_scores = (const float*)d_in[5];
    const float* pos_scores = (const float*)d_in[6];
    const float* neg_scores = (const float*)d_in[7];
    const float* igm        = (const float*)d_in[8];
    float*       out        = (float*)d_out;

    const int P = 50;
    const int G = 32;
    const int B = in_sizes[3] / P;                      // 1024
    const int N = (int)(in_sizes[5] / (B > 0 ? B : 1)); // 100000

    multi_loss_kernel<<<dim3(B), dim3(256), 0, stream>>>(
        wacc, wdiv, wfair, pos_items, all_scores, pos_scores, neg_scores,
        igm, out, P, N, G);
}